// GCNModel_45389214384861
// MI455X (gfx1250) — compile-verified
//
#include <hip/hip_runtime.h>
#include <hip/hip_bf16.h>

typedef float v2f __attribute__((ext_vector_type(2)));
typedef float v4f __attribute__((ext_vector_type(4)));
typedef float v8f __attribute__((ext_vector_type(8)));

#define IN_F  128
#define HID_F 128
#define OUT_F 64

// ---------------------------------------------------------------------------
// Degree / normalization kernels
// ---------------------------------------------------------------------------
__global__ __launch_bounds__(256) void deg_init_kernel(float* deg, int n) {
    int i = blockIdx.x * 256 + threadIdx.x;
    if (i < n) deg[i] = 1.0f;                 // self-loop contributes 1
}

__global__ __launch_bounds__(256) void deg_count_kernel(const int* __restrict__ dst,
                                                        float* deg, int e) {
    int i = blockIdx.x * 256 + threadIdx.x;
    if (i < e) atomicAdd(&deg[dst[i]], 1.0f);
}

__global__ __launch_bounds__(256) void dinv_kernel(float* deg, int n) {
    int i = blockIdx.x * 256 + threadIdx.x;
    if (i < n) deg[i] = rsqrtf(deg[i]);       // deg >= 1 always (self-loop)
}

// ---------------------------------------------------------------------------
// WMMA GEMM:  OUT[M x NOUT] = f(X[M x 128]) @ W[128 x NOUT]
// f = identity, or relu(x + bias_in[k]) fused into the LDS staging load.
// Uses V_WMMA_F32_16X16X4_F32 (exact fp32, matches reference precision).
// Block = 256 threads = 8 waves. NOUT=128: 1 row-strip x 8 col tiles.
//                                NOUT=64 : 2 row-strips x 4 col tiles.
// M must be a multiple of ROWS (100000 % 32 == 0).
// ---------------------------------------------------------------------------
template<int NOUT, bool FUSE_RELU_BIAS>
__global__ __launch_bounds__(256) void gcn_gemm_wmma(
    const float* __restrict__ X, const float* __restrict__ W,
    const float* __restrict__ bias_in, float* __restrict__ OUT)
{
    constexpr int K         = 128;
    constexpr int COL_TILES = NOUT / 16;        // waves per row strip
    constexpr int ROW_STRIPS= 8 / COL_TILES;
    constexpr int ROWS      = ROW_STRIPS * 16;
    constexpr int LDSS      = 132;              // padded stride: bank-conflict free
    __shared__ float Xs[ROWS * LDSS];

    const int tid      = threadIdx.x;
    const int rowblock = blockIdx.x * ROWS;

    // Cooperative float4 staging of the X tile (ROWS x 128) into LDS.
    constexpr int VECS = ROWS * K / 4;
    for (int v = tid; v < VECS; v += 256) {
        const int r  = v / (K / 4);
        const int c4 = (v % (K / 4)) * 4;
        v4f d = *(const v4f*)(X + (size_t)(rowblock + r) * K + c4);
        if constexpr (FUSE_RELU_BIAS) {
            v4f b = *(const v4f*)(bias_in + c4);
            d.x = fmaxf(d.x + b.x, 0.0f);
            d.y = fmaxf(d.y + b.y, 0.0f);
            d.z = fmaxf(d.z + b.z, 0.0f);
            d.w = fmaxf(d.w + b.w, 0.0f);
        }
        *(v4f*)(&Xs[r * LDSS + c4]) = d;
    }
    __syncthreads();

    const int wave    = tid >> 5;
    const int lane    = tid & 31;
    const int strip   = wave / COL_TILES;
    const int colbase = (wave % COL_TILES) * 16;
    const int m       = lane & 15;              // A: row within tile; B: column
    const int khalf   = (lane >> 4) * 2;        // 0 for lanes 0-15, 2 for 16-31

    const float* xrow = &Xs[(strip * 16 + m) * LDSS];

    v8f acc = {};
#pragma unroll
    for (int k0 = 0; k0 < K; k0 += 4) {
        // A fragment (16x4 f32): lane<16 holds {K0,K1}, lane>=16 holds {K2,K3}
        v2f a = *(const v2f*)(xrow + k0 + khalf);
        // B fragment (4x16 f32): lane%16 = column n, lane-half selects K pair
        v2f b;
        b.x = W[(size_t)(k0 + khalf + 0) * NOUT + colbase + m];
        b.y = W[(size_t)(k0 + khalf + 1) * NOUT + colbase + m];
        acc = __builtin_amdgcn_wmma_f32_16x16x4_f32(
            false, a, false, b, (short)0, acc, false, false);
    }

    // D layout: VGPR r -> M = r + 8*(lane>>4), N = lane&15
    const int rbase = rowblock + strip * 16 + ((lane >> 4) << 3);
    const int col   = colbase + m;
#pragma unroll
    for (int r = 0; r < 8; ++r)
        OUT[(size_t)(rbase + r) * NOUT + col] = acc[r];
}

// ---------------------------------------------------------------------------
// Aggregation: init pass writes self-loop term (+ optional bias), scatter pass
// does the edge-wise normalized scatter-add with f32 global atomics.
// ---------------------------------------------------------------------------
template<int F, bool ADD_BIAS>
__global__ __launch_bounds__(256) void agg_init_kernel(
    const float* __restrict__ H, const float* __restrict__ dinv,
    const float* __restrict__ bias, float* __restrict__ out, int n)
{
    long long i = (long long)blockIdx.x * 256 + threadIdx.x;
    if (i >= (long long)n * F) return;
    const int node = (int)(i / F);
    const int f    = (int)(i % F);
    const float di = dinv[node];
    float v = di * di * H[i];
    if constexpr (ADD_BIAS) v += bias[f];
    out[i] = v;
}

template<int F>
__global__ __launch_bounds__(256) void agg_scatter_kernel(
    const int* __restrict__ src, const int* __restrict__ dst,
    const float* __restrict__ dinv, const float* __restrict__ H,
    float* __restrict__ out, int nedges)
{
    constexpr int TPE = F / 4;                  // threads per edge (float4 each)
    long long tid = (long long)blockIdx.x * 256 + threadIdx.x;
    int e = (int)(tid / TPE);
    if (e >= nedges) return;
    const int sub = (int)(tid % TPE);
    const int s = src[e];
    const int d = dst[e];
    const float norm = dinv[s] * dinv[d];
    const int f = sub * 4;
    v4f msg = *(const v4f*)(H + (size_t)s * F + f);
    float* o = out + (size_t)d * F + f;
    atomicAdd(o + 0, norm * msg.x);
    atomicAdd(o + 1, norm * msg.y);
    atomicAdd(o + 2, norm * msg.z);
    atomicAdd(o + 3, norm * msg.w);
}

// ---------------------------------------------------------------------------
extern "C" void kernel_launch(void* const* d_in, const int* in_sizes, int n_in,
                              void* d_out, int out_size, void* d_ws, size_t ws_size,
                              hipStream_t stream)
{
    const float* x   = (const float*)d_in[0];
    const int*  edge = (const int*)d_in[1];     // [2, E] (per harness int mapping)
    const float* W1  = (const float*)d_in[2];
    const float* b1  = (const float*)d_in[3];
    const float* W2  = (const float*)d_in[4];
    const float* b2  = (const float*)d_in[5];
    float* out = (float*)d_out;

    const int n = in_sizes[0] / IN_F;           // 100000
    const int e = in_sizes[1] / 2;              // 1600000
    const int* srcI = edge;
    const int* dstI = edge + e;

    // Workspace layout: dinv | bufA (n*128: H1, later T2) | bufB (n*128: acc1)
    float* dinv = (float*)d_ws;
    size_t off  = ((size_t)n * sizeof(float) + 255) & ~(size_t)255;
    float* bufA = (float*)((char*)d_ws + off);
    float* bufB = (float*)((char*)d_ws + off + (size_t)n * HID_F * sizeof(float));

    const int nblk  = (n + 255) / 256;
    const int eblk  = (e + 255) / 256;

    // 1) symmetric normalization factors
    deg_init_kernel <<<nblk, 256, 0, stream>>>(dinv, n);
    deg_count_kernel<<<eblk, 256, 0, stream>>>(dstI, dinv, e);
    dinv_kernel     <<<nblk, 256, 0, stream>>>(dinv, n);

    // 2) H1 = x @ W1
    gcn_gemm_wmma<HID_F, false><<<n / 16, 256, 0, stream>>>(x, W1, nullptr, bufA);

    // 3) acc1 = dinv^2*H1 ; acc1[dst] += dinv[s]*dinv[d]*H1[src]
    agg_init_kernel<HID_F, false>
        <<<(int)(((long long)n * HID_F + 255) / 256), 256, 0, stream>>>(
            bufA, dinv, nullptr, bufB, n);
    agg_scatter_kernel<HID_F>
        <<<(int)(((long long)e * (HID_F / 4) + 255) / 256), 256, 0, stream>>>(
            srcI, dstI, dinv, bufA, bufB, e);

    // 4) T2 = relu(acc1 + b1) @ W2   (relu+bias fused into LDS staging)
    gcn_gemm_wmma<OUT_F, true><<<n / 32, 256, 0, stream>>>(bufB, W2, b1, bufA);

    // 5) out = b2 + dinv^2*T2 ; out[dst] += dinv[s]*dinv[d]*T2[src]
    agg_init_kernel<OUT_F, true>
        <<<(int)(((long long)n * OUT_F + 255) / 256), 256, 0, stream>>>(
            bufA, dinv, b2, out, n);
    agg_scatter_kernel<OUT_F>
        <<<(int)(((long long)e * (OUT_F / 4) + 255) / 256), 256, 0, stream>>>(
            srcI, dstI, dinv, bufA, out, e);
}